// IntergraphInteractV2_33560874451729
// MI455X (gfx1250) — compile-verified
//
#include <hip/hip_runtime.h>
#include <hip/hip_bf16.h>

#define DIMF 64

typedef float v2f __attribute__((ext_vector_type(2)));
typedef float v8f __attribute__((ext_vector_type(8)));

// order-preserving float -> uint encoding (for atomicMax on floats incl. negatives)
static __device__ __forceinline__ unsigned enc_f32(float f) {
  unsigned u = __float_as_uint(f);
  return (u >> 31) ? ~u : (u | 0x80000000u);
}
static __device__ __forceinline__ float dec_f32(unsigned u) {
  return (u >> 31) ? __uint_as_float(u & 0x7FFFFFFFu) : __uint_as_float(~u);
}
#define ENC_NEG_INF 0x007FFFFFu  // enc(-inf)

// ---------------------------------------------------------------------------
// K0: copy Xq -> out, Xt -> out (Xt region), zero accumulators, init max/sum
// ---------------------------------------------------------------------------
__global__ void igx_init_copy(const float* __restrict__ Xq,
                              const float* __restrict__ Xt,
                              float* __restrict__ outXq,
                              float* __restrict__ XtOut,
                              float* __restrict__ Uacc,
                              float* __restrict__ sArr,
                              unsigned* __restrict__ mEnc,
                              int nXt, int nXq, int NT) {
  int i = blockIdx.x * blockDim.x + threadIdx.x;
  if (i < nXt) { XtOut[i] = Xt[i]; Uacc[i] = 0.0f; }
  if (i < nXq) { outXq[i] = Xq[i]; }
  if (i < NT)  { sArr[i] = 0.0f; mEnc[i] = ENC_NEG_INF; }
}

// ---------------------------------------------------------------------------
// K1: consensus overwrite Xt_c[vc[j]] = Xq[uc[j]]
// ---------------------------------------------------------------------------
__global__ void igx_consensus(const float* __restrict__ Xq,
                              const int* __restrict__ uc,
                              const int* __restrict__ vc,
                              float* __restrict__ XtOut, int M) {
  int i = blockIdx.x * blockDim.x + threadIdx.x;
  if (i >= M * DIMF) return;
  int j = i >> 6, d = i & 63;
  XtOut[(size_t)vc[j] * DIMF + d] = Xq[(size_t)uc[j] * DIMF + d];
}

// ---------------------------------------------------------------------------
// K2: Yq = Xq @ W  for W_alpha and W_beta, via v_wmma_f32_16x16x4_f32.
// Block = 128 threads = 4 waves; wave w computes the 16x16 output tile at
// (m0 = 16*blockIdx.x, n0 = 16*w). blockIdx.y selects alpha/beta.
// EXEC is all-ones (exact grid, no divergence) as WMMA requires.
// A (16x4 f32, 2 VGPR): lanes 0-15 hold K={0,1}, lanes 16-31 hold K={2,3}.
// B (4x16 f32, 2 VGPR): lanes 0-15 hold K rows {0,1}, lanes 16-31 rows {2,3}.
// C (16x16 f32, 8 VGPR): vgpr r: lanes 0-15 -> M=r, lanes 16-31 -> M=8+r.
// ---------------------------------------------------------------------------
__global__ void igx_yq_gemm(const float* __restrict__ Xq,
                            const float* __restrict__ Wa,
                            const float* __restrict__ Wb,
                            float* __restrict__ Yqa,
                            float* __restrict__ Yqb) {
  const int lane = threadIdx.x & 31;
  const int wave = threadIdx.x >> 5;        // 0..3 -> N tile
  const int m0 = blockIdx.x * 16;
  const int n0 = wave * 16;
  const float* __restrict__ W = blockIdx.y ? Wb : Wa;
  float* __restrict__ Y       = blockIdx.y ? Yqb : Yqa;

  const int half = lane >> 4;               // 0 -> K pair {0,1}; 1 -> {2,3}
  const int l16  = lane & 15;

  v8f acc = {};
#pragma unroll
  for (int k = 0; k < DIMF; k += 4) {
    v2f a, b;
    a.x = Xq[(size_t)(m0 + l16) * DIMF + k + 2 * half + 0];
    a.y = Xq[(size_t)(m0 + l16) * DIMF + k + 2 * half + 1];
    b.x = W[(size_t)(k + 2 * half + 0) * DIMF + n0 + l16];
    b.y = W[(size_t)(k + 2 * half + 1) * DIMF + n0 + l16];
    acc = __builtin_amdgcn_wmma_f32_16x16x4_f32(
        /*neg_a=*/false, a, /*neg_b=*/false, b,
        /*c_mod=*/(short)0, acc, /*reuse_a=*/false, /*reuse_b=*/false);
  }
#pragma unroll
  for (int r = 0; r < 8; ++r) {
    int row = m0 + r + 8 * half;
    Y[(size_t)row * DIMF + n0 + l16] = acc[r];
  }
}

// ---------------------------------------------------------------------------
// K3 (pass A): per edge, alpha = elu(Yqa[u].XtC[v] + bA), beta = sigmoid(...)
// store both; segment atomic-max of alpha (encoded) per target v.
// One lane per edge; rows are 256B-aligned float4 streams in L2.
// ---------------------------------------------------------------------------
__global__ void igx_edge_score(const int* __restrict__ u_idx,
                               const int* __restrict__ v_idx,
                               const float* __restrict__ Yqa,
                               const float* __restrict__ Yqb,
                               const float* __restrict__ XtC,
                               const float* __restrict__ bA,
                               const float* __restrict__ bB,
                               float* __restrict__ alphaArr,
                               float* __restrict__ betaArr,
                               unsigned* __restrict__ mEnc, int E) {
  int e = blockIdx.x * blockDim.x + threadIdx.x;
  if (e >= E) return;
  int u = u_idx[e];
  int v = v_idx[e];
  const float4* __restrict__ ya = (const float4*)(Yqa + (size_t)u * DIMF);
  const float4* __restrict__ yb = (const float4*)(Yqb + (size_t)u * DIMF);
  const float4* __restrict__ xv = (const float4*)(XtC + (size_t)v * DIMF);
  float da = 0.0f, db = 0.0f;
#pragma unroll
  for (int i = 0; i < DIMF / 4; ++i) {
    float4 t = xv[i];
    float4 p = ya[i];
    float4 q = yb[i];
    da += p.x * t.x + p.y * t.y + p.z * t.z + p.w * t.w;
    db += q.x * t.x + q.y * t.y + q.z * t.z + q.w * t.w;
  }
  float ar = da + bA[0];
  float alpha = (ar > 0.0f) ? ar : (__expf(ar) - 1.0f);   // ELU
  float br = db + bB[0];
  float beta = 1.0f / (1.0f + __expf(-br));               // sigmoid
  alphaArr[e] = alpha;
  betaArr[e]  = beta;
  atomicMax(&mEnc[v], enc_f32(alpha));
}

// ---------------------------------------------------------------------------
// K4 (pass B): 16 lanes per edge (4 floats each).
// w = exp(alpha - m[v]); h = (1-beta)*xq + beta*xt;
// atomic: s[v] += w (lane 0);  U[v*64 + 4*sub .. +3] += w*h   (4 atomics/lane)
// ---------------------------------------------------------------------------
__global__ void igx_edge_scatter(const int* __restrict__ u_idx,
                                 const int* __restrict__ v_idx,
                                 const float* __restrict__ Xq,
                                 const float* __restrict__ XtC,
                                 const float* __restrict__ alphaArr,
                                 const float* __restrict__ betaArr,
                                 const unsigned* __restrict__ mEnc,
                                 float* __restrict__ sArr,
                                 float* __restrict__ Uacc, int E) {
  long long tid = (long long)blockIdx.x * blockDim.x + threadIdx.x;
  int e   = (int)(tid >> 4);
  int sub = (int)(tid & 15);
  if (e >= E) return;
  int u = u_idx[e];
  int v = v_idx[e];
  float alpha = alphaArr[e];
  float beta  = betaArr[e];
  float w = __expf(alpha - dec_f32(mEnc[v]));
  if (sub == 0) atomicAdd(&sArr[v], w);
  float4 xq = *(const float4*)(Xq  + (size_t)u * DIMF + sub * 4);
  float4 xt = *(const float4*)(XtC + (size_t)v * DIMF + sub * 4);
  float ob = 1.0f - beta;
  float* Uv = Uacc + (size_t)v * DIMF + sub * 4;
  atomicAdd(Uv + 0, w * (ob * xq.x + beta * xt.x));
  atomicAdd(Uv + 1, w * (ob * xq.y + beta * xt.y));
  atomicAdd(Uv + 2, w * (ob * xq.z + beta * xt.z));
  atomicAdd(Uv + 3, w * (ob * xq.w + beta * xt.w));
}

// ---------------------------------------------------------------------------
// K5: finalize rows with at least one edge: XtOut[v,:] = U[v,:] / s[v]
// (rows with s==0 keep the Xt_c contents already written by K0/K1)
// ---------------------------------------------------------------------------
__global__ void igx_finalize(const float* __restrict__ Uacc,
                             const float* __restrict__ sArr,
                             float* __restrict__ XtOut, int nXt) {
  int i = blockIdx.x * blockDim.x + threadIdx.x;
  if (i >= nXt) return;
  int r = i >> 6;
  float sv = sArr[r];
  if (sv > 0.0f) XtOut[i] = Uacc[i] / sv;
}

// ---------------------------------------------------------------------------
extern "C" void kernel_launch(void* const* d_in, const int* in_sizes, int n_in,
                              void* d_out, int out_size, void* d_ws, size_t ws_size,
                              hipStream_t stream) {
  const float* Xq   = (const float*)d_in[0];
  const float* Xt   = (const float*)d_in[1];
  const float* Wa   = (const float*)d_in[2];
  const float* bA   = (const float*)d_in[3];
  const float* Wb   = (const float*)d_in[4];
  const float* bB   = (const float*)d_in[5];
  const int*  u_idx = (const int*)d_in[6];
  const int*  v_idx = (const int*)d_in[7];
  const int*  uc    = (const int*)d_in[8];
  const int*  vc    = (const int*)d_in[9];

  const int NQ = in_sizes[0] / DIMF;   // 2048
  const int NT = in_sizes[1] / DIMF;   // 131072
  const int E  = in_sizes[6];          // 2,000,000
  const int M  = in_sizes[8];          // 1024
  const int nXq = NQ * DIMF;
  const int nXt = NT * DIMF;

  float* outXq = (float*)d_out;
  float* XtOut = (float*)d_out + (size_t)nXq;

  // workspace carve-out (256B aligned)
  char* ws = (char*)d_ws;
  size_t off = 0;
  auto take = [&](size_t bytes) -> void* {
    void* p = ws + off;
    off = (off + bytes + 255) & ~(size_t)255;
    return p;
  };
  float*    Yqa      = (float*)take((size_t)nXq * 4);
  float*    Yqb      = (float*)take((size_t)nXq * 4);
  float*    alphaArr = (float*)take((size_t)E * 4);
  float*    betaArr  = (float*)take((size_t)E * 4);
  float*    sArr     = (float*)take((size_t)NT * 4);
  unsigned* mEnc     = (unsigned*)take((size_t)NT * 4);
  float*    Uacc     = (float*)take((size_t)nXt * 4);
  (void)ws_size; (void)n_in; (void)out_size;

  const int B = 256;

  // K0: copies + zero accumulators
  igx_init_copy<<<(nXt + B - 1) / B, B, 0, stream>>>(
      Xq, Xt, outXq, XtOut, Uacc, sArr, mEnc, nXt, nXq, NT);

  // K1: consensus overwrite
  igx_consensus<<<(M * DIMF + B - 1) / B, B, 0, stream>>>(Xq, uc, vc, XtOut, M);

  // K2: WMMA GEMMs Yq = Xq @ W  (grid.y: 0=alpha, 1=beta)
  dim3 ggrid(NQ / 16, 2, 1);
  igx_yq_gemm<<<ggrid, 128, 0, stream>>>(Xq, Wa, Wb, Yqa, Yqb);

  // K3: per-edge scores + segment max
  igx_edge_score<<<(E + B - 1) / B, B, 0, stream>>>(
      u_idx, v_idx, Yqa, Yqb, XtOut, bA, bB, alphaArr, betaArr, mEnc, E);

  // K4: exp-weighted scatter accumulate (16 lanes / edge)
  long long totB = (long long)E * 16;
  igx_edge_scatter<<<(unsigned)((totB + B - 1) / B), B, 0, stream>>>(
      u_idx, v_idx, Xq, XtOut, alphaArr, betaArr, mEnc, sArr, Uacc, E);

  // K5: finalize
  igx_finalize<<<(nXt + B - 1) / B, B, 0, stream>>>(Uacc, sArr, XtOut, nXt);
}